// ScaleShiftInvariantLoss_20641612824745
// MI455X (gfx1250) — compile-verified
//
#include <hip/hip_runtime.h>
#include <hip/hip_bf16.h>

// ---------------------------------------------------------------------------
// ScaleShiftInvariantLoss for (64,1,480,640) f32 pred/target + bool mask.
// Memory-bound two-pass reduction:
//   pass1: per-sample moments (n, Σp, Σt, Σp², Σpt)   -> per-block partials
//   ab   : WMMA column-sum of partials -> per-sample (a,b,n)
//   pass2: per-sample Σ|a·p+b−t|·m                    -> per-block partials
//   fin  : WMMA column-sum + scalar combine -> loss
// 177MB/pass; pass-1 footprint fits the 192MB L2 so pass-2 re-reads from L2.
// All reductions are fixed-order (shfl_xor / WMMA / serial) => deterministic.
// Partials padded to 76 rows (multiple of 4) so the WMMA reduce loop has no
// bounds guards: each B operand is one unconditional global_load_b64.
// ---------------------------------------------------------------------------

#define NS   307200       // elements per sample (480*640)
#define NV   76800        // float4 vectors per sample
#define PB   75           // blocks per sample   (NV / PB = 1024 vec per block)
#define PBP  76           // padded rows (multiple of 4); row 75 zero-filled
#define BVEC 1024
#define TPB  256
#define ITERS 4           // BVEC / TPB
#define EPS  1e-6f

typedef float f4  __attribute__((ext_vector_type(4)));
typedef float v2f __attribute__((ext_vector_type(2)));
typedef float v8f __attribute__((ext_vector_type(8)));

// workspace layout (float offsets); total = 5*64*76 + 192 + 64*76 ≈ 115 KB
#define OFF_P1 0
#define OFF_AB (5 * 64 * PBP)
#define OFF_P2 (OFF_AB + 3 * 64)

__device__ __forceinline__ float wave_sum(float v) {
#pragma unroll
  for (int off = 16; off > 0; off >>= 1) v += __shfl_xor(v, off, 32);
  return v;
}

// --------------------------- pass 1: moments -------------------------------
__global__ __launch_bounds__(TPB) void pass1(const float* __restrict__ pred,
                                             const float* __restrict__ target,
                                             const unsigned char* __restrict__ mask,
                                             float* __restrict__ p1) {
  const int s   = blockIdx.y;
  const int blk = blockIdx.x;
  const size_t base = (size_t)s * NS;
  const f4* p4 = reinterpret_cast<const f4*>(pred + base);
  const f4* t4 = reinterpret_cast<const f4*>(target + base);
  const unsigned int* m4 = reinterpret_cast<const unsigned int*>(mask + base);

  float cn = 0.f, sp = 0.f, st = 0.f, sp2 = 0.f, spt = 0.f;
  const int v0 = blk * BVEC + threadIdx.x;
#pragma unroll
  for (int i = 0; i < ITERS; ++i) {
    const int v = v0 + i * TPB;
    const f4 p = p4[v];
    const f4 t = t4[v];
    const unsigned int mw = m4[v];
#pragma unroll
    for (int c = 0; c < 4; ++c) {
      const float pe = (c == 0) ? p.x : (c == 1) ? p.y : (c == 2) ? p.z : p.w;
      const float te = (c == 0) ? t.x : (c == 1) ? t.y : (c == 2) ? t.z : t.w;
      const bool m = (((mw >> (8 * c)) & 0xFFu) != 0u) &&
                     __builtin_isfinite(pe) && __builtin_isfinite(te);
      const float pm = m ? pe : 0.f;
      const float tm = m ? te : 0.f;
      cn += m ? 1.f : 0.f;
      sp += pm;
      st += tm;
      sp2 = fmaf(pm, pm, sp2);
      spt = fmaf(pm, tm, spt);
    }
  }

  __shared__ float red[TPB / 32][5];
  const int wave = threadIdx.x >> 5, lane = threadIdx.x & 31;
  cn = wave_sum(cn); sp = wave_sum(sp); st = wave_sum(st);
  sp2 = wave_sum(sp2); spt = wave_sum(spt);
  if (lane == 0) {
    red[wave][0] = cn; red[wave][1] = sp; red[wave][2] = st;
    red[wave][3] = sp2; red[wave][4] = spt;
  }
  __syncthreads();
  if (threadIdx.x == 0) {
    float r[5] = {0.f, 0.f, 0.f, 0.f, 0.f};
    for (int w = 0; w < TPB / 32; ++w)
      for (int q = 0; q < 5; ++q) r[q] += red[w][q];
    const size_t o = (size_t)s * PBP + blk;
    for (int q = 0; q < 5; ++q) p1[(size_t)q * (64 * PBP) + o] = r[q];
    if (blk == 0) {  // zero the pad row so the WMMA reduce needs no guards
      for (int q = 0; q < 5; ++q)
        p1[(size_t)q * (64 * PBP) + (size_t)s * PBP + PB] = 0.f;
    }
  }
}

// ---------------- WMMA columnwise sum of a [PBP x 16] tile -----------------
// base[col*PBP + r]; A = ones(16x4) so D[m][n] = sum_k B[k][n] + C.
// Layout-robust: with A = ones the result is invariant to the k<->slot
// permutation of the f32 B layout; column n's four B slots are
// {VGPR0,VGPR1} x {lane n, lane n+16}, filled here with 4 distinct rows via
// one contiguous 8-byte load per lane per step (rows r0, r0+1).
__device__ __forceinline__ float colsum16_wmma(const float* __restrict__ base,
                                               int lane) {
  const int col = lane & 15;
  const int half = lane >> 4;
  v2f ones = {1.f, 1.f};
  v8f c = {0.f, 0.f, 0.f, 0.f, 0.f, 0.f, 0.f, 0.f};
  const float* colp = base + (size_t)col * PBP + 2 * half;  // 8B aligned
#pragma unroll 4
  for (int k0 = 0; k0 < PBP; k0 += 4) {
    const v2f b = *reinterpret_cast<const v2f*>(colp + k0);  // rows r0, r0+1
    c = __builtin_amdgcn_wmma_f32_16x16x4_f32(false, ones, false, b,
                                              (short)0, c, false, false);
  }
  return c[0];  // lane n (and n+16) -> column-sum of column (lane&15)
}

// ------------------- per-sample (a,b,n) from partials ----------------------
__global__ void kernel_ab(const float* __restrict__ p1, float* __restrict__ ab) {
  __shared__ float sums[5][64];
  const int lane = threadIdx.x;  // 32 threads, one full wave (EXEC all ones)
  for (int q = 0; q < 5; ++q) {
    for (int g = 0; g < 4; ++g) {
      const float cs = colsum16_wmma(
          p1 + (size_t)q * (64 * PBP) + (size_t)(g * 16) * PBP, lane);
      if (lane < 16) sums[q][g * 16 + lane] = cs;
    }
  }
  __syncthreads();
  for (int s = lane; s < 64; s += 32) {
    const float n = sums[0][s], sp = sums[1][s], st = sums[2][s];
    const float sp2 = sums[3][s], spt = sums[4][s];
    const float det = n * sp2 - sp * sp;
    const bool safe = fabsf(det) > EPS;
    const float dets = safe ? det : 1.f;
    const float a = safe ? (n * spt - sp * st) / dets : 1.f;
    const float b = safe ? (st - a * sp) / fmaxf(n, 1.f) : 0.f;
    ab[3 * s + 0] = a;
    ab[3 * s + 1] = b;
    ab[3 * s + 2] = n;
  }
}

// --------------------------- pass 2: residual ------------------------------
__global__ __launch_bounds__(TPB) void pass2(const float* __restrict__ pred,
                                             const float* __restrict__ target,
                                             const unsigned char* __restrict__ mask,
                                             const float* __restrict__ ab,
                                             float* __restrict__ p2) {
  const int s   = blockIdx.y;
  const int blk = blockIdx.x;
  const float a = ab[3 * s + 0];
  const float b = ab[3 * s + 1];
  const size_t base = (size_t)s * NS;
  const f4* p4 = reinterpret_cast<const f4*>(pred + base);
  const f4* t4 = reinterpret_cast<const f4*>(target + base);
  const unsigned int* m4 = reinterpret_cast<const unsigned int*>(mask + base);

  float sr = 0.f;
  const int v0 = blk * BVEC + threadIdx.x;
#pragma unroll
  for (int i = 0; i < ITERS; ++i) {
    const int v = v0 + i * TPB;
    const f4 p = p4[v];
    const f4 t = t4[v];
    const unsigned int mw = m4[v];
#pragma unroll
    for (int c = 0; c < 4; ++c) {
      const float pe = (c == 0) ? p.x : (c == 1) ? p.y : (c == 2) ? p.z : p.w;
      const float te = (c == 0) ? t.x : (c == 1) ? t.y : (c == 2) ? t.z : t.w;
      const bool m = (((mw >> (8 * c)) & 0xFFu) != 0u) &&
                     __builtin_isfinite(pe) && __builtin_isfinite(te);
      const float pm = m ? pe : 0.f;
      const float tm = m ? te : 0.f;
      const float mf = m ? 1.f : 0.f;
      sr += fabsf(fmaf(a, pm, b) - tm) * mf;
    }
  }

  __shared__ float red[TPB / 32];
  const int wave = threadIdx.x >> 5, lane = threadIdx.x & 31;
  sr = wave_sum(sr);
  if (lane == 0) red[wave] = sr;
  __syncthreads();
  if (threadIdx.x == 0) {
    float r = 0.f;
    for (int w = 0; w < TPB / 32; ++w) r += red[w];
    p2[(size_t)s * PBP + blk] = r;
    if (blk == 0) p2[(size_t)s * PBP + PB] = 0.f;  // zero pad row
  }
}

// ------------------------------- finalize ----------------------------------
__global__ void finalize(const float* __restrict__ p2,
                         const float* __restrict__ ab,
                         float* __restrict__ out) {
  __shared__ float rs[64];
  const int lane = threadIdx.x;  // 32 threads, one full wave
  for (int g = 0; g < 4; ++g) {
    const float cs = colsum16_wmma(p2 + (size_t)(g * 16) * PBP, lane);
    if (lane < 16) rs[g * 16 + lane] = cs;
  }
  __syncthreads();
  if (lane == 0) {
    float acc = 0.f, cnt = 0.f;
    for (int s = 0; s < 64; ++s) {
      const float n = ab[3 * s + 2];
      const float per = rs[s] / fmaxf(n, 1.f);
      const float inc = (n >= 2.f) ? 1.f : 0.f;
      acc += per * inc;
      cnt += inc;
    }
    out[0] = (cnt > 0.f) ? acc / fmaxf(cnt, 1.f) : 0.f;
  }
}

// ------------------------------- launcher ----------------------------------
extern "C" void kernel_launch(void* const* d_in, const int* in_sizes, int n_in,
                              void* d_out, int out_size, void* d_ws, size_t ws_size,
                              hipStream_t stream) {
  const float* pred            = (const float*)d_in[0];
  const float* target          = (const float*)d_in[1];
  const unsigned char* mask    = (const unsigned char*)d_in[2];  // jnp.bool_: 1 byte/elem
  float* ws  = (float*)d_ws;                                     // needs ~115 KB
  float* out = (float*)d_out;

  dim3 grid(PB, 64);
  pass1<<<grid, TPB, 0, stream>>>(pred, target, mask, ws + OFF_P1);
  kernel_ab<<<1, 32, 0, stream>>>(ws + OFF_P1, ws + OFF_AB);
  pass2<<<grid, TPB, 0, stream>>>(pred, target, mask, ws + OFF_AB, ws + OFF_P2);
  finalize<<<1, 32, 0, stream>>>(ws + OFF_P2, ws + OFF_AB, out);
}